// LastAttenion_49847390437916
// MI455X (gfx1250) — compile-verified
//
#include <hip/hip_runtime.h>
#include <math.h>
#include <stdint.h>

// Problem constants (B == S == HEADS == 48, H = 768, D = 16)
#define B_     48
#define S_     48
#define HEADS_ 48
#define H_     768
#define D_     16

typedef __attribute__((ext_vector_type(2))) float        v2f;
typedef __attribute__((ext_vector_type(8))) float        v8f;
typedef __attribute__((ext_vector_type(4))) unsigned int v4u;
typedef __attribute__((ext_vector_type(4))) int          v4i;
typedef __attribute__((ext_vector_type(8))) int          v8i;

// Detect TDM builtin (device pass only; host pass parses the fallback path).
#ifndef HAVE_TDM
#  if defined(__HIP_DEVICE_COMPILE__) && defined(__has_builtin)
#    if __has_builtin(__builtin_amdgcn_tensor_load_to_lds)
#      define HAVE_TDM 1
#    endif
#  endif
#endif
#ifndef HAVE_TDM
#  define HAVE_TDM 0
#endif

// Definitive confirmation that the emitted device code uses the TDM path.
#if defined(__HIP_DEVICE_COMPILE__) && !HAVE_TDM
#  error "gfx1250 TDM builtin (__builtin_amdgcn_tensor_load_to_lds) not available"
#endif

// ---------------------------------------------------------------------------
// GEMM: Y[m,n] = sum_k X[m,k]*W[n,k] + bias[n]   (K = N = 768, M runtime)
// WG = 128 threads (4 waves, wave32), WG tile 32(M) x 64(N), waves in 2x2:
// each wave owns a 16x32 block => two f32 WMMA accumulators sharing one A
// fragment (3 LDS dwords per v_wmma_f32_16x16x4_f32 instead of 4).
// LDS staged in 64-wide K chunks, stride 65 (64-bank conflict free).
// TDM path: double-buffered tensor_load_to_lds with hardware pad (interval
// 64 dwords, amount 1 dword => stride 65), overlapped with compute via
// s_wait_tensorcnt(2).
// ---------------------------------------------------------------------------
#define GK   768
#define GN   768
#define BK   64
#define TM   32
#define TN   64
#define LSTR (BK + 1)

#if HAVE_TDM
// 2D tile load, Global -> LDS via Tensor Data Mover.
//  tile_x elements per row (contiguous), tile_y rows, row stride in elements,
//  tensor_{x,y} = in-bounds extent from the tile origin (OOB reads -> 0).
__device__ __forceinline__ void tdm_load_2d(unsigned lds_off, const void* gptr,
                                            unsigned tile_x, unsigned tile_y,
                                            unsigned tensor_x, unsigned tensor_y,
                                            unsigned row_stride_elems) {
  const unsigned long long ga = (unsigned long long)(uintptr_t)gptr;
  v4u g0;
  g0[0] = 1u;                                   // count=1 user descriptor
  g0[1] = lds_off;                              // lds_addr (bytes)
  g0[2] = (unsigned)(ga & 0xFFFFFFFFu);         // global_addr[31:0]
  g0[3] = (unsigned)((ga >> 32) & 0x01FFFFFFu)  // global_addr[56:32]
          | (2u << 30);                         // type = 2 ("image")
  v8i g1;
  g1[0] = (int)((2u << 16)                      // data_size = 4 bytes
                | (1u << 20)                    // pad_enable
                | (5u << 22));                  // pad_interval: 64 dwords
                                                // pad_amount = 0 => 1 dword
  g1[1] = (int)((tensor_x & 0xFFFFu) << 16);                    // tensor_dim0[15:0]
  g1[2] = (int)(((tensor_x >> 16) & 0xFFFFu) |
                ((tensor_y & 0xFFFFu) << 16));                  // dim0 hi | dim1 lo
  g1[3] = (int)(((tensor_y >> 16) & 0xFFFFu) |
                ((tile_x & 0xFFFFu) << 16));                    // dim1 hi | tile_dim0
  g1[4] = (int)(tile_y & 0xFFFFu);                              // tile_dim1 (tile_dim2=0)
  g1[5] = (int)row_stride_elems;                                // tensor_dim0_stride[31:0]
  g1[6] = 0;                                                    // stride0 hi | stride1 lo
  g1[7] = 0;
  const v4i z4 = {0, 0, 0, 0};
#if __clang_major__ >= 23
  const v8i z8 = {0, 0, 0, 0, 0, 0, 0, 0};
  __builtin_amdgcn_tensor_load_to_lds(g0, g1, z4, z4, z8, 0);
#else
  __builtin_amdgcn_tensor_load_to_lds(g0, g1, z4, z4, 0);
#endif
}
#endif  // HAVE_TDM

__global__ void __launch_bounds__(128)
gemm_bias_wmma(const float* __restrict__ X,
               const float* __restrict__ W,
               const float* __restrict__ bias,
               float* __restrict__ Y, int M) {
#if HAVE_TDM
  __shared__ float lds_x[2][TM * LSTR];
  __shared__ float lds_w[2][TN * LSTR];
#else
  __shared__ float lds_x[1][TM * LSTR];
  __shared__ float lds_w[1][TN * LSTR];
#endif

  const int m0   = blockIdx.x * TM;
  const int n0   = blockIdx.y * TN;
  const int tid  = threadIdx.x;
  const int wave = tid >> 5;
  const int lane = tid & 31;
  const int half = lane >> 4;   // 0: lanes 0-15, 1: lanes 16-31
  const int lrow = lane & 15;
  const int mi   = wave >> 1;   // M sub-tile (0..1)
  const int ni   = wave & 1;    // N sub-tile (0..1)

  v8f acc0 = {};
  v8f acc1 = {};

#if HAVE_TDM
  const unsigned xrows = (unsigned)((M - m0) < TM ? (M - m0) : TM);
  if (wave == 0) {
    tdm_load_2d((unsigned)(uintptr_t)&lds_x[0][0], X + (size_t)m0 * GK,
                BK, TM, BK, xrows, GK);
    tdm_load_2d((unsigned)(uintptr_t)&lds_w[0][0], W + (size_t)n0 * GK,
                BK, TN, BK, TN, GK);
  }
#pragma unroll 1
  for (int c = 0; c < GK / BK; ++c) {
    const int buf = c & 1;
    if (wave == 0) {
      if (c + 1 < GK / BK) {
        const int k1 = (c + 1) * BK;
        tdm_load_2d((unsigned)(uintptr_t)&lds_x[buf ^ 1][0],
                    X + (size_t)m0 * GK + k1, BK, TM, BK, xrows, GK);
        tdm_load_2d((unsigned)(uintptr_t)&lds_w[buf ^ 1][0],
                    W + (size_t)n0 * GK + k1, BK, TN, BK, TN, GK);
        __builtin_amdgcn_s_wait_tensorcnt(2);  // chunk c's pair complete
      } else {
        __builtin_amdgcn_s_wait_tensorcnt(0);
      }
    }
    __syncthreads();
    const float* lx = &lds_x[buf][0];
    const float* lw = &lds_w[buf][0];
#pragma unroll
    for (int kk = 0; kk < BK; kk += 4) {
      v2f a, b0, b1;
      a.x  = lx[(mi * 16 + lrow) * LSTR + kk + 2 * half + 0];
      a.y  = lx[(mi * 16 + lrow) * LSTR + kk + 2 * half + 1];
      b0.x = lw[(ni * 32 + lrow) * LSTR + kk + 2 * half + 0];
      b0.y = lw[(ni * 32 + lrow) * LSTR + kk + 2 * half + 1];
      b1.x = lw[(ni * 32 + 16 + lrow) * LSTR + kk + 2 * half + 0];
      b1.y = lw[(ni * 32 + 16 + lrow) * LSTR + kk + 2 * half + 1];
      acc0 = __builtin_amdgcn_wmma_f32_16x16x4_f32(false, a, false, b0,
                                                   (short)0, acc0, false, false);
      acc1 = __builtin_amdgcn_wmma_f32_16x16x4_f32(false, a, false, b1,
                                                   (short)0, acc1, false, false);
    }
    __syncthreads();
  }
#else
#pragma unroll 1
  for (int k0 = 0; k0 < GK; k0 += BK) {
    for (int i = tid; i < TM * BK; i += 128) {
      int r = i >> 6, cc = i & (BK - 1);
      int gr = m0 + r;
      if (gr >= M) gr = M - 1;  // clamp; out-of-range rows never stored
      lds_x[0][r * LSTR + cc] = X[(size_t)gr * GK + k0 + cc];
    }
    for (int i = tid; i < TN * BK; i += 128) {
      int r = i >> 6, cc = i & (BK - 1);
      lds_w[0][r * LSTR + cc] = W[(size_t)(n0 + r) * GK + k0 + cc];
    }
    __syncthreads();
    const float* lx = &lds_x[0][0];
    const float* lw = &lds_w[0][0];
#pragma unroll
    for (int kk = 0; kk < BK; kk += 4) {
      v2f a, b0, b1;
      a.x  = lx[(mi * 16 + lrow) * LSTR + kk + 2 * half + 0];
      a.y  = lx[(mi * 16 + lrow) * LSTR + kk + 2 * half + 1];
      b0.x = lw[(ni * 32 + lrow) * LSTR + kk + 2 * half + 0];
      b0.y = lw[(ni * 32 + lrow) * LSTR + kk + 2 * half + 1];
      b1.x = lw[(ni * 32 + 16 + lrow) * LSTR + kk + 2 * half + 0];
      b1.y = lw[(ni * 32 + 16 + lrow) * LSTR + kk + 2 * half + 1];
      acc0 = __builtin_amdgcn_wmma_f32_16x16x4_f32(false, a, false, b0,
                                                   (short)0, acc0, false, false);
      acc1 = __builtin_amdgcn_wmma_f32_16x16x4_f32(false, a, false, b1,
                                                   (short)0, acc1, false, false);
    }
    __syncthreads();
  }
#endif

  // C/D layout: VGPR r holds row (r + 8*half), column lrow of each 16x16 tile
  const int col0 = n0 + ni * 32 + lrow;
  const int col1 = col0 + 16;
  const float bv0 = bias[col0];
  const float bv1 = bias[col1];
#pragma unroll
  for (int r = 0; r < 8; ++r) {
    const int row = m0 + mi * 16 + 8 * half + r;
    if (row < M) {
      Y[(size_t)row * GN + col0] = acc0[r] + bv0;
      Y[(size_t)row * GN + col1] = acc1[r] + bv1;
    }
  }
}

// ---------------------------------------------------------------------------
// scores[b,h,j] = dot16(q0[b,h,:], q1full[b*S+j, h*16 : h*16+16])
// alpha1[b,h,j] = softmax_j( sigmoid(scores) ).  One block per (b,h).
// ---------------------------------------------------------------------------
__global__ void __launch_bounds__(64)
scores_softmax_kernel(const float* __restrict__ q0,
                      const float* __restrict__ q1,
                      float* __restrict__ alpha1) {
  const int b = blockIdx.x;
  const int h = blockIdx.y;
  const int tid = threadIdx.x;
  __shared__ float sh_q0[D_];
  __shared__ float red[64];

  if (tid < D_) sh_q0[tid] = q0[(size_t)b * H_ + h * D_ + tid];
  __syncthreads();

  float e = 0.f;
  if (tid < S_) {
    const float4* row4 = (const float4*)(q1 + (size_t)(b * S_ + tid) * H_ + h * D_);
    const float4* q4   = (const float4*)sh_q0;
    float s = 0.f;
#pragma unroll
    for (int d = 0; d < D_ / 4; ++d) {
      float4 r = row4[d];
      float4 q = q4[d];
      s += q.x * r.x + q.y * r.y + q.z * r.z + q.w * r.w;
    }
    float sig = 1.f / (1.f + expf(-s));
    e = expf(sig);  // sigmoid in (0,1): exp safe without max-subtract
  }
  red[tid] = e;
  __syncthreads();
  for (int off = 32; off > 0; off >>= 1) {
    if (tid < off) red[tid] += red[tid + off];
    __syncthreads();
  }
  if (tid < S_) alpha1[(size_t)(b * HEADS_ + h) * S_ + tid] = e / red[0];
}

// ---------------------------------------------------------------------------
// alpha2[b,h,k] = softmax over h of 2*alpha1[b,h,k].  One block per (b,k).
// ---------------------------------------------------------------------------
__global__ void __launch_bounds__(64)
head_softmax_kernel(const float* __restrict__ alpha1,
                    float* __restrict__ alpha2) {
  const int b = blockIdx.x;
  const int k = blockIdx.y;
  const int tid = threadIdx.x;
  __shared__ float red[64];

  float e = 0.f;
  if (tid < HEADS_) e = expf(2.f * alpha1[(size_t)(b * HEADS_ + tid) * S_ + k]);
  red[tid] = e;
  __syncthreads();
  for (int off = 32; off > 0; off >>= 1) {
    if (tid < off) red[tid] += red[tid + off];
    __syncthreads();
  }
  if (tid < HEADS_) alpha2[(size_t)(b * HEADS_ + tid) * S_ + k] = e / red[0];
}

// ---------------------------------------------------------------------------
// out_alpha[b,h,s,k] = mask[b,s] > 0 ? alpha2[b,h,k] : NaN   (21 MB store)
// float4-vectorized: k fastest (48 = 12 x float4), all bases 16B aligned.
// ---------------------------------------------------------------------------
__global__ void __launch_bounds__(256)
broadcast_alpha_kernel(const float* __restrict__ alpha2,
                       const float* __restrict__ mask,
                       float* __restrict__ out_alpha) {
  const size_t total4 = (size_t)B_ * HEADS_ * S_ * (S_ / 4);
  size_t idx = (size_t)blockIdx.x * blockDim.x + threadIdx.x;
  if (idx >= total4) return;
  int k4 = (int)(idx % (S_ / 4));
  size_t t = idx / (S_ / 4);
  int s = (int)(t % S_);
  t /= S_;
  int h = (int)(t % HEADS_);
  int b = (int)(t / HEADS_);
  float4 v = *(const float4*)(alpha2 + (size_t)(b * HEADS_ + h) * S_ + 4 * k4);
  if (!(mask[b * S_ + s] > 0.f)) {
    const float qn = __builtin_nanf("");
    v.x = qn; v.y = qn; v.z = qn; v.w = qn;
  }
  ((float4*)out_alpha)[idx] = v;
}

// ---------------------------------------------------------------------------
// a[b, k*16+d] = sum_h alpha2[b,h,k] * (sum_s mask[b,s]*q2full[h*S+k, s*16+d])
// One thread per (b,k,d): 36,864 threads, L2-resident reads.
// ---------------------------------------------------------------------------
__global__ void __launch_bounds__(256)
final_a_kernel(const float* __restrict__ alpha2,
               const float* __restrict__ q2,
               const float* __restrict__ mask,
               float* __restrict__ a) {
  const int idx = blockIdx.x * blockDim.x + threadIdx.x;
  if (idx >= B_ * S_ * D_) return;
  const int d = idx & (D_ - 1);
  const int k = (idx >> 4) % S_;
  const int b = idx / (S_ * D_);

  float acc = 0.f;
  for (int h = 0; h < HEADS_; ++h) {
    const float w = alpha2[(size_t)(b * HEADS_ + h) * S_ + k];
    const float* row = q2 + (size_t)(h * S_ + k) * H_ + d;
    float t = 0.f;
#pragma unroll 4
    for (int s = 0; s < S_; ++s) t += mask[b * S_ + s] * row[s * D_];
    acc += w * t;
  }
  a[(size_t)b * H_ + k * D_ + d] = acc;
}

// ---------------------------------------------------------------------------
extern "C" void kernel_launch(void* const* d_in, const int* in_sizes, int n_in,
                              void* d_out, int out_size, void* d_ws, size_t ws_size,
                              hipStream_t stream) {
  (void)in_sizes; (void)n_in; (void)out_size; (void)ws_size;

  const float* ht1    = (const float*)d_in[0];
  const float* hidden = (const float*)d_in[1];
  const float* mask   = (const float*)d_in[2];
  const float* W0     = (const float*)d_in[3];
  const float* b0     = (const float*)d_in[4];
  const float* W1     = (const float*)d_in[5];
  const float* b1     = (const float*)d_in[6];
  const float* W2     = (const float*)d_in[7];
  const float* b2     = (const float*)d_in[8];

  float* ws     = (float*)d_ws;
  float* q0     = ws;                                // 48*768    = 36,864
  float* q1     = q0 + (size_t)B_ * H_;              // 2304*768  = 1,769,472
  float* q2     = q1 + (size_t)B_ * S_ * H_;         // 2304*768  = 1,769,472
  float* alpha1 = q2 + (size_t)B_ * S_ * H_;         // 48*48*48  = 110,592
  float* alpha2 = alpha1 + (size_t)B_ * HEADS_ * S_; // 48*48*48  = 110,592

  float* out_a     = (float*)d_out;                  // (a, alpha) in return order
  float* out_alpha = out_a + (size_t)B_ * H_;

  // Three GEMMs (fp32 WMMA + TDM staging): q0 (48x768), q1 & q2 (2304x768)
  gemm_bias_wmma<<<dim3((B_ + TM - 1) / TM, GN / TN), dim3(128), 0, stream>>>(
      ht1, W0, b0, q0, B_);
  gemm_bias_wmma<<<dim3((B_ * S_) / TM, GN / TN), dim3(128), 0, stream>>>(
      hidden, W1, b1, q1, B_ * S_);
  gemm_bias_wmma<<<dim3((B_ * S_) / TM, GN / TN), dim3(128), 0, stream>>>(
      hidden, W2, b2, q2, B_ * S_);

  // Attention weights
  scores_softmax_kernel<<<dim3(B_, HEADS_), dim3(64), 0, stream>>>(q0, q1, alpha1);
  head_softmax_kernel<<<dim3(B_, S_), dim3(64), 0, stream>>>(alpha1, alpha2);

  // Outputs
  const size_t total4 = (size_t)B_ * HEADS_ * S_ * (S_ / 4);
  broadcast_alpha_kernel<<<(unsigned)((total4 + 255) / 256), dim3(256), 0, stream>>>(
      alpha2, mask, out_alpha);
  final_a_kernel<<<(B_ * S_ * D_ + 255) / 256, dim3(256), 0, stream>>>(
      alpha2, q2, mask, out_a);
}